// BackProjNet_61881888800891
// MI455X (gfx1250) — compile-verified
//
#include <hip/hip_runtime.h>

typedef __attribute__((ext_vector_type(2))) float v2f;
typedef __attribute__((ext_vector_type(8))) float v8f;
typedef __attribute__((ext_vector_type(4))) float f4;
typedef __attribute__((ext_vector_type(4))) int   i4;

#define TPV        1024          // terms per voxel = VIEWS*EXTENT
#define NVOX       65536         // NX*NY
#define VOX_PER_BLK 16
#define SCALE_F    3.0679615757712823e-3f   // (2*pi)/(2*VIEWS*EXTENT) = pi/1024

// 512 threads = 16 wave32 per block; wave w handles voxel blockIdx.x*16 + w.
// Each voxel's 1024 terms are contiguous in idx/weight -> fully coalesced
// 128-bit non-temporal streaming loads. x (1.47MB) is L2-resident -> cheap gathers.
__global__ __launch_bounds__(512)
void backproj_gather_wmma(const float* __restrict__ x,
                          const float* __restrict__ w,
                          const float* __restrict__ bias,
                          const int*   __restrict__ idx,
                          float*       __restrict__ out) {
    __shared__ float P[VOX_PER_BLK][33];   // padded rows -> spread LDS banks

    const int wave = threadIdx.x >> 5;
    const int lane = threadIdx.x & 31;
    const int vox  = (int)blockIdx.x * VOX_PER_BLK + wave;

    const i4* __restrict__ idx4 = (const i4*)(idx + (size_t)vox * TPV);
    const f4* __restrict__ w4   = (const f4*)(w   + (size_t)vox * TPV);

    // 1024 terms / (32 lanes * 4-wide) = 8 iterations; 4 independent accumulators.
    float a0 = 0.f, a1 = 0.f, a2 = 0.f, a3 = 0.f;
    #pragma unroll
    for (int it = 0; it < 8; ++it) {
        const int off = it * 32 + lane;                    // coalesced across wave
        const i4 iv = __builtin_nontemporal_load(idx4 + off);  // read-once stream
        const f4 wv = __builtin_nontemporal_load(w4   + off);  // read-once stream
        a0 = fmaf(x[iv.x], wv.x, a0);                      // L2-resident gathers
        a1 = fmaf(x[iv.y], wv.y, a1);
        a2 = fmaf(x[iv.z], wv.z, a2);
        a3 = fmaf(x[iv.w], wv.w, a3);
    }
    P[wave][lane] = (a0 + a1) + (a2 + a3);
    __syncthreads();

    // Wave 0 reduces all 16 voxels' 32 partials with chained f32 WMMA, B = ones:
    //   D[m][n] = sum_k A[m][k] * 1 + C[m][n]  -> after 8 chunks of K=4,
    //   column 0 holds the exact f32 sum of 32 partials for each voxel row m.
    if (threadIdx.x < 32) {
        const int row   = lane & 15;          // A-matrix M = lane % 16
        const int kbase = (lane >> 4) * 2;    // lanes 0-15: K={0,1}; 16-31: K={2,3}
        v8f acc = {};
        const v2f bones = {1.0f, 1.0f};
        #pragma unroll
        for (int c = 0; c < 8; ++c) {
            v2f A;
            A.x = P[row][c * 4 + kbase + 0];
            A.y = P[row][c * 4 + kbase + 1];
            // 8 args: (neg_a, A, neg_b, B, c_mod, C, reuse_a, reuse_b)
            acc = __builtin_amdgcn_wmma_f32_16x16x4_f32(
                false, A, false, bones, (short)0, acc, false, false);
        }
        // C/D layout: lane 0 holds (N=0, M=0..7) in v0..v7; lane 16 holds M=8..15.
        if (lane == 0 || lane == 16) {
            const int mbase = (lane == 0) ? 0 : 8;
            #pragma unroll
            for (int r = 0; r < 8; ++r) {
                const int v = (int)blockIdx.x * VOX_PER_BLK + mbase + r;
                // flip(axis 2,3) on 256x256 == flat index reversal
                out[(NVOX - 1) - v] = acc[r] * SCALE_F + bias[v];
            }
        }
    }
}

extern "C" void kernel_launch(void* const* d_in, const int* in_sizes, int n_in,
                              void* d_out, int out_size, void* d_ws, size_t ws_size,
                              hipStream_t stream) {
    // setup_inputs order: x (f32), weight (f32), bias (f32), indices (int)
    const float* x    = (const float*)d_in[0];
    const float* wgt  = (const float*)d_in[1];
    const float* bias = (const float*)d_in[2];
    const int*   idx  = (const int*)  d_in[3];
    float*       out  = (float*)d_out;

    const int blocks = NVOX / VOX_PER_BLK;   // 4096 blocks * 512 threads
    backproj_gather_wmma<<<blocks, 512, 0, stream>>>(x, wgt, bias, idx, out);
}